// YOLOV5_40046275068427
// MI455X (gfx1250) — compile-verified
//
#include <hip/hip_runtime.h>
#include <stdint.h>

#define BATCH   16
#define NANCH   25200
#define NCLS    80
#define KSEL    4096
#define MAXDET  1000
#define CONF_T  0.55f
#define IOU_T   0.45f
#define MAX_WH  7680.0f
#define IMGSZ   640.0f
#define NBUCK   65536
#define MASKW   (KSEL / 32)   // 128 words of suppression bits per box

typedef float v8f __attribute__((ext_vector_type(8)));
typedef float v2f __attribute__((ext_vector_type(2)));

// ---- order-preserving float <-> uint key (descending sort on uint) ----
__device__ __forceinline__ uint32_t f2key(float f) {
  uint32_t u = __float_as_uint(f);
  return u ^ ((u & 0x80000000u) ? 0xFFFFFFFFu : 0x80000000u);
}
__device__ __forceinline__ float key2f(uint32_t u) {
  uint32_t b = (u & 0x80000000u) ? (u ^ 0x80000000u) : ~u;
  return __uint_as_float(b);
}

// ===== Stage 1: per-anchor confidence / class argmax (bandwidth bound) =====
__global__ void yolo_score(const float* __restrict__ pred,
                           uint32_t* __restrict__ key, int* __restrict__ cls) {
  int gid = blockIdx.x * blockDim.x + threadIdx.x;
  if (gid >= BATCH * NANCH) return;
  const float* p = pred + (size_t)gid * (5 + NCLS);
  __builtin_prefetch(p + (5 + NCLS) * 64, 0, 1);   // global_prefetch_b8
  float obj = p[4];
  float best = -1e30f; int bi = 0;
  #pragma unroll 4
  for (int c = 0; c < NCLS; ++c) {
    float v = p[5 + c];
    if (v > best) { best = v; bi = c; }
  }
  float conf = best * obj;
  bool valid = (obj > CONF_T) && (conf > CONF_T);
  key[gid] = f2key(valid ? conf : -1.0f);
  cls[gid] = bi;
}

// ===== Stage 2: exact top-K threshold via two-level 16-bit histograms =====
__global__ void yolo_hist1(const uint32_t* __restrict__ key, uint32_t* __restrict__ hist) {
  int gid = blockIdx.x * blockDim.x + threadIdx.x;
  if (gid >= BATCH * NANCH) return;
  int b = gid / NANCH;
  atomicAdd(&hist[(size_t)b * NBUCK + (key[gid] >> 16)], 1u);
}

__global__ void yolo_scan1(const uint32_t* __restrict__ hist, int* __restrict__ meta) {
  int b = blockIdx.x, t = threadIdx.x;
  const uint32_t* h = hist + (size_t)b * NBUCK;
  const int CH = NBUCK / 256;
  __shared__ uint32_t part[256], pre[256];
  uint32_t s = 0;
  for (int q = 0; q < CH; ++q) s += h[NBUCK - 1 - (t * CH + q)];
  part[t] = s; __syncthreads();
  if (t == 0) { uint32_t r = 0; for (int x = 0; x < 256; ++x) { pre[x] = r; r += part[x]; } }
  __syncthreads();
  uint32_t run = pre[t];
  for (int q = 0; q < CH; ++q) {
    int bucket = NBUCK - 1 - (t * CH + q);
    uint32_t c = h[bucket];
    if (run < KSEL && run + c >= KSEL) { meta[b * 16 + 0] = bucket; meta[b * 16 + 1] = (int)run; }
    run += c;
  }
}

__global__ void yolo_hist2(const uint32_t* __restrict__ key, const int* __restrict__ meta,
                           uint32_t* __restrict__ hist) {
  int gid = blockIdx.x * blockDim.x + threadIdx.x;
  if (gid >= BATCH * NANCH) return;
  int b = gid / NANCH;
  uint32_t k = key[gid];
  if ((int)(k >> 16) == meta[b * 16 + 0])
    atomicAdd(&hist[(size_t)b * NBUCK + (k & 0xFFFFu)], 1u);
}

__global__ void yolo_scan2(const uint32_t* __restrict__ hist, int* __restrict__ meta) {
  int b = blockIdx.x, t = threadIdx.x;
  const uint32_t* h = hist + (size_t)b * NBUCK;
  int target = KSEL - meta[b * 16 + 1];
  const int CH = NBUCK / 256;
  __shared__ uint32_t part[256], pre[256];
  uint32_t s = 0;
  for (int q = 0; q < CH; ++q) s += h[NBUCK - 1 - (t * CH + q)];
  part[t] = s; __syncthreads();
  if (t == 0) { uint32_t r = 0; for (int x = 0; x < 256; ++x) { pre[x] = r; r += part[x]; } }
  __syncthreads();
  uint32_t run = pre[t];
  for (int q = 0; q < CH; ++q) {
    int bucket = NBUCK - 1 - (t * CH + q);
    uint32_t c = h[bucket];
    if ((int)run < target && (int)(run + c) >= target) {
      meta[b * 16 + 2] = bucket;               // low-16 of threshold key
      meta[b * 16 + 3] = target - (int)run;    // quota from equal-key group
    }
    run += c;
  }
}

__global__ void yolo_compact(const uint32_t* __restrict__ key, int* meta,
                             unsigned long long* __restrict__ topk) {
  int gid = blockIdx.x * blockDim.x + threadIdx.x;
  if (gid >= BATCH * NANCH) return;
  int b = gid / NANCH, i = gid % NANCH;
  uint32_t k = key[gid];
  uint32_t thr = ((uint32_t)meta[b * 16 + 0] << 16) | (uint32_t)meta[b * 16 + 2];
  int pos = -1;
  if (k > thr) pos = atomicAdd(&meta[b * 16 + 4], 1);
  else if (k == thr) {
    int q = atomicAdd(&meta[b * 16 + 5], 1);
    if (q < meta[b * 16 + 3]) pos = atomicAdd(&meta[b * 16 + 4], 1);
  }
  if (pos >= 0 && pos < KSEL)
    topk[(size_t)b * KSEL + pos] =
        ((unsigned long long)k << 32) | (uint32_t)(~(uint32_t)i);  // tie-break: lower idx first
}

// ===== Stage 3: in-LDS bitonic sort of 4096 (key|~idx), gather boxes =====
__global__ void __launch_bounds__(512)
yolo_sortgather(unsigned long long* __restrict__ topk, const float* __restrict__ pred,
                const int* __restrict__ cls, float4* __restrict__ boxes,
                float* __restrict__ area, float* __restrict__ scr, float* __restrict__ clst) {
  __shared__ unsigned long long sk[KSEL];   // 32 KB
  int b = blockIdx.x, t = threadIdx.x;
  for (int r = t; r < KSEL; r += blockDim.x) sk[r] = topk[(size_t)b * KSEL + r];
  for (int size = 2; size <= KSEL; size <<= 1) {
    for (int stride = size >> 1; stride > 0; stride >>= 1) {
      __syncthreads();
      for (int idx = t; idx < KSEL / 2; idx += blockDim.x) {
        int i = (idx << 1) - (idx & (stride - 1));
        int j = i + stride;
        unsigned long long a = sk[i], c = sk[j];
        bool descSeg = ((i & size) == 0);
        bool sw = descSeg ? (a < c) : (a > c);
        if (sw) { sk[i] = c; sk[j] = a; }
      }
    }
  }
  __syncthreads();
  for (int r = t; r < KSEL; r += blockDim.x) {
    unsigned long long kv = sk[r];
    uint32_t k = (uint32_t)(kv >> 32);
    uint32_t i = ~(uint32_t)kv;
    const float* p = pred + ((size_t)b * NANCH + i) * (5 + NCLS);
    float cx = p[0], cy = p[1], w = p[2], h = p[3];
    float4 bx = make_float4(cx - 0.5f * w, cy - 0.5f * h, cx + 0.5f * w, cy + 0.5f * h);
    size_t o = (size_t)b * KSEL + r;
    boxes[o] = bx;
    area[o]  = (bx.z - bx.x) * (bx.w - bx.y);
    scr[o]   = key2f(k);
    clst[o]  = (float)cls[(size_t)b * NANCH + i];
  }
}

// ===== Stage 4: suppression bitmask. Each wave: 16 suppressors x 32 victims.
// Pair-sum of areas (denominator term) via V_WMMA_F32_16X16X4_F32; intersection
// on VALU; bits packed with wave32 ballot. Boxes staged to LDS with CDNA5
// async-to-LDS loads. =====
__global__ void __launch_bounds__(128)
yolo_mask(const float4* __restrict__ boxes, const float* __restrict__ area,
          const float* __restrict__ clst, uint32_t* __restrict__ mask) {
  int i0 = blockIdx.x * 64;        // suppressor rows handled by this block
  int w  = blockIdx.y;             // victim word (32 victims)
  int j0 = w * 32;
  int b  = blockIdx.z;
  int t  = threadIdx.x;

  if (j0 + 31 <= i0) {             // need victim j > suppressor s: impossible here
    if (t < 64) mask[((size_t)b * KSEL + i0 + t) * MASKW + w] = 0u;
    return;
  }

  __shared__ float4 sb[64];        // suppressor boxes (class-shifted)
  __shared__ float4 vb[32];        // victim boxes (class-shifted)
  __shared__ float  sa[64], va[32];

  // --- async stage 96 float4 box records straight into LDS ---
  if (t < 96) {
    uint32_t lds  = (t < 64) ? (uint32_t)(uintptr_t)&sb[t]
                             : (uint32_t)(uintptr_t)&vb[t - 64];
    uint32_t goff = (t < 64) ? (uint32_t)(((size_t)b * KSEL + i0 + t) * 16)
                             : (uint32_t)(((size_t)b * KSEL + j0 + (t - 64)) * 16);
    asm volatile("global_load_async_to_lds_b128 %0, %1, %2"
                 :: "v"(lds), "v"(goff), "s"(boxes) : "memory");
  }
  asm volatile("s_wait_asynccnt 0" ::: "memory");
  __syncthreads();

  // apply per-class coordinate offset, stage areas
  if (t < 64) {
    float sh = clst[(size_t)b * KSEL + i0 + t] * MAX_WH;
    float4 v = sb[t]; v.x += sh; v.y += sh; v.z += sh; v.w += sh; sb[t] = v;
    sa[t] = area[(size_t)b * KSEL + i0 + t];
  } else if (t < 96) {
    int j = t - 64;
    float sh = clst[(size_t)b * KSEL + j0 + j] * MAX_WH;
    float4 v = vb[j]; v.x += sh; v.y += sh; v.z += sh; v.w += sh; vb[j] = v;
    va[j] = area[(size_t)b * KSEL + j0 + j];
  }
  __syncthreads();

  int wv = t >> 5, lane = t & 31;
  int sbase = wv * 16;             // this wave's 16 suppressor rows

  // A (16x4): K0 = area_s, K1 = 1, K2/K3 = 0  (lanes 16-31 hold K2,K3)
  v2f a;
  a.x = (lane < 16) ? sa[sbase + lane] : 0.0f;
  a.y = (lane < 16) ? 1.0f : 0.0f;
  v8f cz = {};

  uint32_t balLo[8], balHi[8];
  #pragma unroll
  for (int hh = 0; hh < 2; ++hh) {
    // B (4x16): row0 = ones, row1 = area_v
    v2f bb;
    bb.x = (lane < 16) ? 1.0f : 0.0f;
    bb.y = (lane < 16) ? va[hh * 16 + lane] : 0.0f;
    // D[m][n] = area_s[m] + area_v[n]   (matrix pipe)
    v8f d = __builtin_amdgcn_wmma_f32_16x16x4_f32(false, a, false, bb,
                                                  (short)0, cz, false, false);
    int col = hh * 16 + (lane & 15);
    float4 vbx = vb[col];
    int jg = j0 + col;
    #pragma unroll
    for (int r = 0; r < 8; ++r) {
      int m = r + ((lane >= 16) ? 8 : 0);   // C/D layout: VGPR r -> rows r, r+8
      int sl = sbase + m;
      float4 sbx = sb[sl];
      float xx1 = fmaxf(sbx.x, vbx.x), yy1 = fmaxf(sbx.y, vbx.y);
      float xx2 = fminf(sbx.z, vbx.z), yy2 = fminf(sbx.w, vbx.w);
      float iw = fmaxf(xx2 - xx1, 0.0f), ih = fmaxf(yy2 - yy1, 0.0f);
      float inter = iw * ih;
      float denom = d[r] - inter + 1e-7f;   // area_s + area_v - inter + eps
      bool cond = (inter > IOU_T * denom) && (jg > i0 + sl);
      uint32_t bal = __builtin_amdgcn_ballot_w32(cond);
      if (hh == 0) balLo[r] = bal; else balHi[r] = bal;
    }
  }
  if (lane == 0) {
    size_t base = ((size_t)b * KSEL + i0 + sbase) * MASKW + w;
    #pragma unroll
    for (int r = 0; r < 8; ++r) {
      uint32_t lo = balLo[r], hi = balHi[r];
      mask[base + (size_t)r * MASKW]       = (lo & 0xFFFFu) | (hi << 16);
      mask[base + (size_t)(r + 8) * MASKW] = (lo >> 16) | (hi & 0xFFFF0000u);
    }
  }
}

// ===== Stage 5: sequential suppression scan (one block per image) =====
__global__ void __launch_bounds__(128)
yolo_nmsscan(const uint32_t* __restrict__ mask, const float* __restrict__ scr,
             int* __restrict__ list) {
  int b = blockIdx.x, t = threadIdx.x;
  __shared__ uint32_t rem[MASKW];
  uint32_t k0 = 0;
  for (int q = 0; q < 32; ++q)
    if (scr[(size_t)b * KSEL + t * 32 + q] > 0.0f) k0 |= (1u << q);
  rem[t] = ~k0;                    // invalid entries start "removed"
  __syncthreads();
  for (int i = 0; i < KSEL; ++i) {
    bool alive = ((rem[i >> 5] >> (i & 31)) & 1u) == 0u;
    __syncthreads();
    if (alive) rem[t] |= mask[((size_t)b * KSEL + i) * MASKW + t];
    __syncthreads();
  }
  if (t == 0) {
    int cnt = 0;
    int* L = list + b * (MAXDET + 1);
    for (int i = 0; i < KSEL && cnt < MAXDET; ++i)
      if (((rem[i >> 5] >> (i & 31)) & 1u) == 0u) L[1 + cnt++] = i;
    L[0] = cnt;
  }
}

// ===== Stage 6: rescale to original image, emit (B, 1000, 6) =====
__global__ void yolo_final(const float4* __restrict__ boxes, const float* __restrict__ scr,
                           const float* __restrict__ clst, const int* __restrict__ list,
                           const int* __restrict__ ih, const int* __restrict__ iw,
                           float* __restrict__ out) {
  int b = blockIdx.x, t = threadIdx.x;
  float h0 = (float)ih[0], w0 = (float)iw[0];
  float gain = fminf(IMGSZ / h0, IMGSZ / w0);
  float padx = (IMGSZ - w0 * gain) * 0.5f;
  float pady = (IMGSZ - h0 * gain) * 0.5f;
  float inv = 1.0f / gain;
  const int* L = list + b * (MAXDET + 1);
  int cnt = L[0];
  for (int r = t; r < MAXDET; r += blockDim.x) {
    float* o = out + ((size_t)b * MAXDET + r) * 6;
    if (r < cnt) {
      int i = L[1 + r];
      float4 bx = boxes[(size_t)b * KSEL + i];
      float s = scr[(size_t)b * KSEL + i];
      o[0] = rintf(fminf(fmaxf((bx.x - padx) * inv, 0.0f), w0));
      o[1] = rintf(fminf(fmaxf((bx.y - pady) * inv, 0.0f), h0));
      o[2] = rintf(fminf(fmaxf((bx.z - padx) * inv, 0.0f), w0));
      o[3] = rintf(fminf(fmaxf((bx.w - pady) * inv, 0.0f), h0));
      o[4] = s;
      o[5] = clst[(size_t)b * KSEL + i];
    } else {
      for (int c = 0; c < 6; ++c) o[c] = 0.0f;
    }
  }
}

extern "C" void kernel_launch(void* const* d_in, const int* in_sizes, int n_in,
                              void* d_out, int out_size, void* d_ws, size_t ws_size,
                              hipStream_t stream) {
  (void)in_sizes; (void)n_in; (void)out_size; (void)ws_size;
  const float* pred = (const float*)d_in[0];
  const int* ih = (const int*)d_in[1];
  const int* iw = (const int*)d_in[2];

  char* ws = (char*)d_ws;
  size_t o = 0;
  auto take = [&](size_t bytes) { size_t r = o; o += (bytes + 255) & ~(size_t)255; return r; };
  size_t A = (size_t)BATCH * NANCH;
  uint32_t* key   = (uint32_t*)(ws + take(A * 4));
  int*      cls   = (int*)     (ws + take(A * 4));
  uint32_t* hist1 = (uint32_t*)(ws + take((size_t)BATCH * NBUCK * 4));
  uint32_t* hist2 = (uint32_t*)(ws + take((size_t)BATCH * NBUCK * 4));
  int*      meta  = (int*)     (ws + take((size_t)BATCH * 16 * 4));
  unsigned long long* topk = (unsigned long long*)(ws + take((size_t)BATCH * KSEL * 8));
  float4*   boxes = (float4*)  (ws + take((size_t)BATCH * KSEL * 16));
  float*    area  = (float*)   (ws + take((size_t)BATCH * KSEL * 4));
  float*    scr   = (float*)   (ws + take((size_t)BATCH * KSEL * 4));
  float*    clst  = (float*)   (ws + take((size_t)BATCH * KSEL * 4));
  uint32_t* mask  = (uint32_t*)(ws + take((size_t)BATCH * KSEL * MASKW * 4));
  int*      list  = (int*)     (ws + take((size_t)BATCH * (MAXDET + 1) * 4));

  hipMemsetAsync(hist1, 0, (size_t)BATCH * NBUCK * 4, stream);
  hipMemsetAsync(hist2, 0, (size_t)BATCH * NBUCK * 4, stream);
  hipMemsetAsync(meta,  0, (size_t)BATCH * 16 * 4, stream);

  const int tb = 256;
  const int nb = (int)((A + tb - 1) / tb);
  yolo_score  <<<nb, tb, 0, stream>>>(pred, key, cls);
  yolo_hist1  <<<nb, tb, 0, stream>>>(key, hist1);
  yolo_scan1  <<<BATCH, 256, 0, stream>>>(hist1, meta);
  yolo_hist2  <<<nb, tb, 0, stream>>>(key, meta, hist2);
  yolo_scan2  <<<BATCH, 256, 0, stream>>>(hist2, meta);
  yolo_compact<<<nb, tb, 0, stream>>>(key, meta, topk);
  yolo_sortgather<<<BATCH, 512, 0, stream>>>(topk, pred, cls, boxes, area, scr, clst);
  yolo_mask   <<<dim3(KSEL / 64, MASKW, BATCH), 128, 0, stream>>>(boxes, area, clst, mask);
  yolo_nmsscan<<<BATCH, 128, 0, stream>>>(mask, scr, list);
  yolo_final  <<<BATCH, 256, 0, stream>>>(boxes, scr, clst, list, ih, iw, (float*)d_out);
}